// HyperLinear_3590592660278
// MI455X (gfx1250) — compile-verified
//
#include <hip/hip_runtime.h>

typedef float v2f __attribute__((ext_vector_type(2)));
typedef float v8f __attribute__((ext_vector_type(8)));

#define SEQ   720
#define PRED  720
#define RANK  16
#define CTX   128
#define HID   64
#define BATCH 256
#define OUTDIM (RANK * (SEQ + PRED))   // 23040
#define APART  (RANK * PRED)           // 11520 (A_delta part of h)
#define PT     (PRED / 16)             // 45 row tiles
#define NT     (SEQ / 16)              // 45 col tiles

// ---------------------------------------------------------------------------
// Kernel 1a: hidden = relu(ctx @ W1^T + b1)   -> t (BATCH, HID)
// ---------------------------------------------------------------------------
__global__ __launch_bounds__(HID)
void hyper_hidden(const float* __restrict__ ctx,  // (B, CTX)
                  const float* __restrict__ W1,   // (HID, CTX)
                  const float* __restrict__ b1,   // (HID)
                  float* __restrict__ t) {        // (B, HID)
  const int b = blockIdx.x;
  const int k = threadIdx.x;
  const float4* crow = (const float4*)(ctx + (size_t)b * CTX);
  const float4* wrow = (const float4*)(W1 + (size_t)k * CTX);
  float acc = b1[k];
#pragma unroll
  for (int c = 0; c < CTX / 4; ++c) {
    float4 cv = crow[c];
    float4 wv = wrow[c];
    acc += cv.x * wv.x + cv.y * wv.y + cv.z * wv.z + cv.w * wv.w;
  }
  t[(size_t)b * HID + k] = fmaxf(acc, 0.0f);
}

// ---------------------------------------------------------------------------
// Kernel 1b: h = t @ W2^T + b2   -> h (BATCH, OUTDIM)
// Each thread caches one W2 row (64 f32) in VGPRs, reuses across 16 batches.
// grid = (OUTDIM/256, BATCH/16), block = 256
// ---------------------------------------------------------------------------
__global__ __launch_bounds__(256)
void hyper_out(const float* __restrict__ t,    // (B, HID)
               const float* __restrict__ W2,   // (OUTDIM, HID)
               const float* __restrict__ b2,   // (OUTDIM)
               float* __restrict__ h) {        // (B, OUTDIM)
  const int o  = blockIdx.x * 256 + threadIdx.x;
  const int b0 = blockIdx.y * 16;
  const float4* wrow = (const float4*)(W2 + (size_t)o * HID);
  float4 w[16];
#pragma unroll
  for (int i = 0; i < 16; ++i) w[i] = wrow[i];
  const float bias = b2[o];
  for (int j = 0; j < 16; ++j) {
    const int b = b0 + j;
    const float4* trow = (const float4*)(t + (size_t)b * HID);
    float acc = bias;
#pragma unroll
    for (int i = 0; i < 16; ++i) {
      float4 tv = trow[i];
      acc += tv.x * w[i].x + tv.y * w[i].y + tv.z * w[i].z + tv.w * w[i].w;
    }
    h[(size_t)b * OUTDIM + o] = acc;
  }
}

// ---------------------------------------------------------------------------
// Kernel 2: W_eff = W + A_delta @ B_delta ; y = W_eff @ x
// One wave32 per (batch, 16-row stripe). Rank-16 update = chain of 4
// V_WMMA_F32_16X16X4_F32 (exact f32 math, matching reference precision).
// W_eff stores are non-temporal: the 531 MB stream must not evict the
// L2-resident reuse set (W: 2 MB, h: 23.6 MB, x: 0.7 MB).
// Layouts per CDNA5 ISA 7.12.2:
//   A 16x4 f32: lane = M (lane&15), elem j -> K = 2*(lane>>4) + j
//   B 4x16 f32: lane = N (lane&15), elem j -> K = 2*(lane>>4) + j
//   C/D 16x16 : lane = N (lane&15), VGPR v -> M = v + 8*(lane>>4)
// ---------------------------------------------------------------------------
__global__ __launch_bounds__(32)
void weff_y(const float* __restrict__ x,    // (B, SEQ)
            const float* __restrict__ W,    // (PRED, SEQ)
            const float* __restrict__ h,    // (B, OUTDIM) workspace
            float* __restrict__ y,          // (B, PRED)
            float* __restrict__ weff) {     // (B, PRED, SEQ)
  const int blk  = blockIdx.x;
  const int b    = blk / PT;
  const int pt   = blk % PT;
  const int p0   = pt * 16;
  const int lane = threadIdx.x;
  const int n    = lane & 15;       // N (or M for the A operand)
  const int hi   = lane >> 4;
  const int kb   = hi * 2;

  // A operand (A_delta[b, p0+m, :]) : h[b, (p0+m)*RANK + r]
  const float* hA = h + (size_t)b * OUTDIM + (size_t)(p0 + n) * RANK + kb;
  v2f a0 = { hA[0],  hA[1]  };   // K = kb+0..1
  v2f a1 = { hA[4],  hA[5]  };   // K = 4+kb..
  v2f a2 = { hA[8],  hA[9]  };
  v2f a3 = { hA[12], hA[13] };

  const float* hB = h + (size_t)b * OUTDIM + APART;  // B_delta (RANK, SEQ)
  const float* xb = x + (size_t)b * SEQ;
  float* wout = weff + (size_t)b * PRED * SEQ;
  const size_t rowbase = (size_t)(p0 + 8 * hi) * SEQ;

  float part[8] = {0.f, 0.f, 0.f, 0.f, 0.f, 0.f, 0.f, 0.f};

  for (int st = 0; st < NT; ++st) {
    const int col = st * 16 + n;

    // B operands: B_delta[b, K, col], K = 4k + kb + {0,1}
    const float* pb = hB + (size_t)kb * SEQ + col;
    v2f bv0 = { pb[0],          pb[SEQ]          };
    v2f bv1 = { pb[4 * SEQ],    pb[5 * SEQ]      };
    v2f bv2 = { pb[8 * SEQ],    pb[9 * SEQ]      };
    v2f bv3 = { pb[12 * SEQ],   pb[13 * SEQ]     };

    // C = base weight tile (L2-resident, regular temporal loads)
    const float* pw = W + rowbase + col;
    v8f c;
#pragma unroll
    for (int v = 0; v < 8; ++v) c[v] = pw[(size_t)v * SEQ];

    // K=16 accumulation chain
    c = __builtin_amdgcn_wmma_f32_16x16x4_f32(false, a0, false, bv0, (short)0, c, false, false);
    c = __builtin_amdgcn_wmma_f32_16x16x4_f32(false, a1, false, bv1, (short)0, c, false, false);
    c = __builtin_amdgcn_wmma_f32_16x16x4_f32(false, a2, false, bv2, (short)0, c, false, false);
    c = __builtin_amdgcn_wmma_f32_16x16x4_f32(false, a3, false, bv3, (short)0, c, false, false);

    // Stream W_eff out non-temporally (coalesced 64B/half-wave per row),
    // accumulate y partials in-register (avoids re-reading 531 MB later).
    const float xv = xb[col];
    float* po = wout + rowbase + col;
#pragma unroll
    for (int v = 0; v < 8; ++v) {
      __builtin_nontemporal_store(c[v], po + (size_t)v * SEQ);
      part[v] += c[v] * xv;
    }
  }

  // Reduce y partials across each 16-lane half (xor butterfly stays in-half)
#pragma unroll
  for (int v = 0; v < 8; ++v) {
    float s = part[v];
    s += __shfl_xor(s, 1, 32);
    s += __shfl_xor(s, 2, 32);
    s += __shfl_xor(s, 4, 32);
    s += __shfl_xor(s, 8, 32);
    part[v] = s;
  }
  if (n == 0) {
#pragma unroll
    for (int v = 0; v < 8; ++v)
      y[(size_t)b * PRED + p0 + v + 8 * hi] = part[v];
  }
}

// ---------------------------------------------------------------------------
extern "C" void kernel_launch(void* const* d_in, const int* in_sizes, int n_in,
                              void* d_out, int out_size, void* d_ws, size_t ws_size,
                              hipStream_t stream) {
  const float* x   = (const float*)d_in[0];  // (256,720,1)
  const float* ctx = (const float*)d_in[1];  // (256,128)
  const float* W   = (const float*)d_in[2];  // (720,720)
  const float* W1  = (const float*)d_in[3];  // (64,128)
  const float* b1  = (const float*)d_in[4];  // (64)
  const float* W2  = (const float*)d_in[5];  // (23040,64)
  const float* b2  = (const float*)d_in[6];  // (23040)

  float* out  = (float*)d_out;
  float* y    = out;                                  // (256,720,1) flat
  float* weff = out + (size_t)BATCH * PRED;           // (256,720,720) flat

  float* h = (float*)d_ws;                            // (256, 23040)
  float* t = h + (size_t)BATCH * OUTDIM;              // (256, 64)

  hipLaunchKernelGGL(hyper_hidden, dim3(BATCH), dim3(HID), 0, stream,
                     ctx, W1, b1, t);
  hipLaunchKernelGGL(hyper_out, dim3(OUTDIM / 256, BATCH / 16), dim3(256), 0, stream,
                     t, W2, b2, h);
  hipLaunchKernelGGL(weff_y, dim3(BATCH * PT), dim3(32), 0, stream,
                     x, W, h, y, weff);
}